// Attention_56513179681502
// MI455X (gfx1250) — compile-verified
//
#include <hip/hip_runtime.h>
#include <hip/hip_bf16.h>
#include <math.h>

// Problem constants (from reference): B=64, S=2048, H=1024
#define Bsz 64
#define Ssz 2048
#define Hsz 1024
#define NCH 8              // S-chunks per batch for the streaming pass
#define SCH (Ssz / NCH)    // 256 s-values per workgroup
#define NWAVES 8           // 256 threads = 8 wave32
#define SPW (SCH / NWAVES) // 32 s-values per wave

typedef __attribute__((ext_vector_type(2))) float v2f;
typedef __attribute__((ext_vector_type(8))) float v8f;

// ---------------------------------------------------------------------------
// C[M,N] = act( A[M,K] @ Bw[N,K]^T )   (both row-major; NT gemm, torch Linear)
// One wave (32 lanes) per 16x16 C tile, V_WMMA_F32_16X16X4_F32 over K.
// A frag (16x4): lanes 0-15 hold K={0,1}, lanes 16-31 hold K={2,3}, row=lid.
// B frag (4x16): v0 holds K=0 (lanes 0-15) / K=2 (lanes 16-31), v1 K=1/K=3,
//                col=lid  -> per-lane float2 from row (tn+lid) of Bw.
// D (16x16): vgpr r -> row tm + r + 8*half, col tn + lid.
// ---------------------------------------------------------------------------
__global__ void __launch_bounds__(32)
wmma_gemm_nt_kernel(const float* __restrict__ A, const float* __restrict__ Bw,
                    float* __restrict__ C, int M, int N, int K, int applyTanh) {
  const int ntn  = N >> 4;
  const int tm   = (blockIdx.x / ntn) << 4;
  const int tn   = (blockIdx.x % ntn) << 4;
  const int lane = threadIdx.x & 31;
  const int half = lane >> 4;
  const int lid  = lane & 15;

  const float* arow = A  + (size_t)(tm + lid) * K + 2 * half;
  const float* brow = Bw + (size_t)(tn + lid) * K + 2 * half;

  v8f c = {};
  for (int k0 = 0; k0 < K; k0 += 4) {
    v2f a = *(const v2f*)(arow + k0);
    v2f b = *(const v2f*)(brow + k0);
    c = __builtin_amdgcn_wmma_f32_16x16x4_f32(false, a, false, b, (short)0, c,
                                              false, false);
  }
#pragma unroll
  for (int r = 0; r < 8; ++r) {
    float v = c[r];
    if (applyTanh) v = tanhf(v);
    C[(size_t)(tm + r + 8 * half) * N + (tn + lid)] = v;
  }
}

// ---------------------------------------------------------------------------
// Streaming fused attention partial: one workgroup = (batch b, S-chunk ch).
// Each wave handles 32 consecutive s values; online softmax + context accum
// held entirely in VGPRs (q frag 32, enc row 32, acc 32 per lane).
// ---------------------------------------------------------------------------
__global__ void __launch_bounds__(256)
attn_partial_kernel(const float* __restrict__ enc,   // [S,B,H]
                    const float* __restrict__ q,     // [B,H]
                    float* __restrict__ scores,      // [B,S]
                    float* __restrict__ part_m,      // [B*NCH]
                    float* __restrict__ part_l,      // [B*NCH]
                    float* __restrict__ part_acc) {  // [B*NCH, H]
  const int b    = blockIdx.x / NCH;
  const int ch   = blockIdx.x % NCH;
  const int tid  = threadIdx.x;
  const int wave = tid >> 5;
  const int lane = tid & 31;

  // lane h-ownership: h = lane*4 + k*128 + {0..3}, k = 0..7  (covers H=1024)
  const float* qrow = q + (size_t)b * Hsz;
  float4 qf[8];
#pragma unroll
  for (int k = 0; k < 8; ++k) qf[k] = *(const float4*)(qrow + lane * 4 + k * 128);

  float m = -INFINITY, l = 0.f;
  float4 acc[8];
#pragma unroll
  for (int k = 0; k < 8; ++k) acc[k] = make_float4(0.f, 0.f, 0.f, 0.f);

  const int s_base = ch * SCH + wave * SPW;
  for (int t = 0; t < SPW; ++t) {
    const int s = s_base + t;
    const float* erow = enc + ((size_t)s * Bsz + b) * Hsz;
    float4 e[8];
#pragma unroll
    for (int k = 0; k < 8; ++k) e[k] = *(const float4*)(erow + lane * 4 + k * 128);

    float dot = 0.f;
#pragma unroll
    for (int k = 0; k < 8; ++k)
      dot += e[k].x * qf[k].x + e[k].y * qf[k].y + e[k].z * qf[k].z +
             e[k].w * qf[k].w;
#pragma unroll
    for (int off = 16; off > 0; off >>= 1) dot += __shfl_xor(dot, off, 32);

    if (lane == 0) scores[(size_t)b * Ssz + s] = dot;

    const float m_new = fmaxf(m, dot);
    const float cf = __expf(m - m_new);     // 0 on first iter (m = -inf)
    const float w  = __expf(dot - m_new);
    l = l * cf + w;
#pragma unroll
    for (int k = 0; k < 8; ++k) {
      acc[k].x = acc[k].x * cf + e[k].x * w;
      acc[k].y = acc[k].y * cf + e[k].y * w;
      acc[k].z = acc[k].z * cf + e[k].z * w;
      acc[k].w = acc[k].w * cf + e[k].w * w;
    }
    m = m_new;
  }

  // --- combine the 8 waves of this workgroup in LDS ---
  __shared__ float wm[NWAVES], wl[NWAVES];
  __shared__ float accs[Hsz];
  for (int h = tid; h < Hsz; h += 256) accs[h] = 0.f;
  if (lane == 0) { wm[wave] = m; wl[wave] = l; }
  __syncthreads();

  float m_wg = wm[0];
#pragma unroll
  for (int i = 1; i < NWAVES; ++i) m_wg = fmaxf(m_wg, wm[i]);
  float l_wg = 0.f;
#pragma unroll
  for (int i = 0; i < NWAVES; ++i) l_wg += wl[i] * __expf(wm[i] - m_wg);

  const float scale = __expf(m - m_wg);
#pragma unroll
  for (int k = 0; k < 8; ++k) {
    const int h = lane * 4 + k * 128;
    atomicAdd(&accs[h + 0], acc[k].x * scale);
    atomicAdd(&accs[h + 1], acc[k].y * scale);
    atomicAdd(&accs[h + 2], acc[k].z * scale);
    atomicAdd(&accs[h + 3], acc[k].w * scale);
  }
  __syncthreads();

  const int pidx = b * NCH + ch;
  if (tid == 0) { part_m[pidx] = m_wg; part_l[pidx] = l_wg; }
  float* pacc = part_acc + (size_t)pidx * Hsz;
  {
    const int h = tid * 4;  // 256 threads * 4 = 1024
    *(float4*)(pacc + h) = *(const float4*)(accs + h);
  }
}

// ---------------------------------------------------------------------------
// Merge per-chunk partials (log-sum-exp), write normalized context into the
// left half of cat[B,2H] and copy dec_out into the right half (fused concat).
// ---------------------------------------------------------------------------
__global__ void __launch_bounds__(256)
attn_combine_kernel(const float* __restrict__ part_m,
                    const float* __restrict__ part_l,
                    const float* __restrict__ part_acc,
                    const float* __restrict__ dec,
                    float* __restrict__ gM, float* __restrict__ gL,
                    float* __restrict__ cat) {
  const int b = blockIdx.x;
  const int tid = threadIdx.x;

  float m_g = part_m[b * NCH];
#pragma unroll
  for (int c = 1; c < NCH; ++c) m_g = fmaxf(m_g, part_m[b * NCH + c]);
  float sc[NCH];
  float l_g = 0.f;
#pragma unroll
  for (int c = 0; c < NCH; ++c) {
    sc[c] = __expf(part_m[b * NCH + c] - m_g);
    l_g += part_l[b * NCH + c] * sc[c];
  }
  if (tid == 0) { gM[b] = m_g; gL[b] = l_g; }
  const float inv = 1.f / l_g;

  for (int h = tid; h < Hsz; h += 256) {
    float v = 0.f;
#pragma unroll
    for (int c = 0; c < NCH; ++c)
      v += part_acc[(size_t)(b * NCH + c) * Hsz + h] * sc[c];
    cat[(size_t)b * 2 * Hsz + h] = v * inv;
    cat[(size_t)b * 2 * Hsz + Hsz + h] = dec[(size_t)b * Hsz + h];
  }
}

__global__ void __launch_bounds__(256)
weights_kernel(const float* __restrict__ scores, const float* __restrict__ gM,
               const float* __restrict__ gL, float* __restrict__ outw) {
  const int i = blockIdx.x * 256 + threadIdx.x;
  if (i < Bsz * Ssz) {
    const int b = i >> 11;  // / Ssz
    outw[i] = __expf(scores[i] - gM[b]) / gL[b];
  }
}

// ---------------------------------------------------------------------------
extern "C" void kernel_launch(void* const* d_in, const int* in_sizes, int n_in,
                              void* d_out, int out_size, void* d_ws,
                              size_t ws_size, hipStream_t stream) {
  const float* dec   = (const float*)d_in[0];   // [B,H]
  const float* enc   = (const float*)d_in[1];   // [S,B,H]
  const float* W_a   = (const float*)d_in[2];   // [H,H]
  const float* W_out = (const float*)d_in[3];   // [H,2H]
  float* out = (float*)d_out;                   // [B*H context | B*S weights]

  float* ws       = (float*)d_ws;               // ~3.4 MB total
  float* q        = ws;                         // B*H        = 65536
  float* scores   = q + Bsz * Hsz;              // B*S        = 131072
  float* part_m   = scores + Bsz * Ssz;         // B*NCH      = 512
  float* part_l   = part_m + Bsz * NCH;         // 512
  float* gM       = part_l + Bsz * NCH;         // 64
  float* gL       = gM + Bsz;                   // 64
  float* part_acc = gL + Bsz;                   // B*NCH*H    = 524288
  float* cat      = part_acc + Bsz * NCH * Hsz; // B*2H       = 131072

  // 1) q = dec_out @ W_a^T      (M=64, N=1024, K=1024), WMMA f32 16x16x4
  wmma_gemm_nt_kernel<<<(Bsz / 16) * (Hsz / 16), 32, 0, stream>>>(
      dec, W_a, q, Bsz, Hsz, Hsz, 0);

  // 2) fused streaming scores + online-softmax context partials (1 enc pass)
  attn_partial_kernel<<<Bsz * NCH, 256, 0, stream>>>(enc, q, scores, part_m,
                                                     part_l, part_acc);

  // 3) merge partials -> cat = [context | dec_out]
  attn_combine_kernel<<<Bsz, 256, 0, stream>>>(part_m, part_l, part_acc, dec,
                                               gM, gL, cat);

  // 4) weights output
  weights_kernel<<<(Bsz * Ssz) / 256, 256, 0, stream>>>(scores, gM, gL,
                                                        out + Bsz * Hsz);

  // 5) context_out = tanh(cat @ W_out^T)  (M=64, N=1024, K=2048), WMMA + tanh
  wmma_gemm_nt_kernel<<<(Bsz / 16) * (Hsz / 16), 32, 0, stream>>>(
      cat, W_out, out, Bsz, Hsz, 2 * Hsz, 1);
}